// GNS_59493886984440
// MI455X (gfx1250) — compile-verified
//
#include <hip/hip_runtime.h>
#include <math.h>

// Problem constants (match reference)
#define DD       128
#define HH       4
#define N_NODES  50000
#define N_EDGES  800000
#define BN_EPS   1e-5f
#define INV_SQRT_C 0.17677669529663687f   // 1/sqrt(32)

typedef __attribute__((ext_vector_type(2))) float v2f;
typedef __attribute__((ext_vector_type(8))) float v8f;

// ---- monotonic uint encoding of f32 so unsigned atomicMax == float max ----
__device__ __forceinline__ unsigned f32_ord(float f) {
  unsigned u = __float_as_uint(f);
  return (u & 0x80000000u) ? ~u : (u | 0x80000000u);
}
__device__ __forceinline__ float ord_f32(unsigned u) {
  unsigned v = (u & 0x80000000u) ? (u & 0x7fffffffu) : ~u;
  return __uint_as_float(v);
}

// =====================================================================
// Kernel 1: QKV + skip GEMM.  D = X(32x128) @ [Wq|Wk|Wv|Wskip](128x512)
// WMMA f32 16x16x4. Block = 256 threads = 8 waves. 2 M-subtiles/block.
// Skip result (x@Wskip+bskip) is written straight into xout so the
// message pass can atomically accumulate on top of it.
// =====================================================================
#define LDSA_STRIDE 132

__global__ __launch_bounds__(256) void node_qkvs_kernel(
    const float* __restrict__ x,
    const float* __restrict__ Wq, const float* __restrict__ bq,
    const float* __restrict__ Wk, const float* __restrict__ bk,
    const float* __restrict__ Wv, const float* __restrict__ bv,
    const float* __restrict__ Ws, const float* __restrict__ bs,
    float* __restrict__ q, float* __restrict__ k, float* __restrict__ v,
    float* __restrict__ xout, int n)
{
  __shared__ float As[32 * LDSA_STRIDE];
  const int tid  = threadIdx.x;
  const int wave = tid >> 5;
  const int lane = tid & 31;
  const int half = lane >> 4;
  const int idx  = lane & 15;
  const int m_base = blockIdx.x * 32;

  // Stage 32x128 X tile (coalesced, rows clamped)
  for (int i = tid; i < 32 * DD; i += 256) {
    int r = i >> 7, c = i & 127;
    int gr = m_base + r;
    As[r * LDSA_STRIDE + c] = (gr < n) ? x[gr * DD + c] : 0.0f;
  }
  __syncthreads();

  const float* Wmat[4] = {Wq, Wk, Wv, Ws};
  const float* Bias[4] = {bq, bk, bv, bs};
  float*       Outp[4] = {q, k, v, xout};

  // wave handles n-tile (wave + 8*j): j selects which weight matrix.
  v8f acc[2][4];
  #pragma unroll
  for (int mt = 0; mt < 2; ++mt)
    #pragma unroll
    for (int j = 0; j < 4; ++j)
      acc[mt][j] = (v8f){0,0,0,0,0,0,0,0};

  const int colw = wave * 16 + idx;   // column inside the 128-wide matrix

  for (int k0 = 0; k0 < DD; k0 += 4) {
    v2f a[2];
    #pragma unroll
    for (int mt = 0; mt < 2; ++mt) {
      a[mt].x = As[(mt * 16 + idx) * LDSA_STRIDE + k0 + 2 * half + 0];
      a[mt].y = As[(mt * 16 + idx) * LDSA_STRIDE + k0 + 2 * half + 1];
    }
    #pragma unroll
    for (int j = 0; j < 4; ++j) {
      const float* W = Wmat[j];
      v2f b;
      b.x = W[(k0 + 2 * half + 0) * DD + colw];
      b.y = W[(k0 + 2 * half + 1) * DD + colw];
      #pragma unroll
      for (int mt = 0; mt < 2; ++mt)
        acc[mt][j] = __builtin_amdgcn_wmma_f32_16x16x4_f32(
            false, a[mt], false, b, (short)0, acc[mt][j], false, false);
    }
  }

  // bias + store (C/D layout: VGPR mm -> row mm + 8*half, lane idx -> col)
  #pragma unroll
  for (int j = 0; j < 4; ++j) {
    float bb = Bias[j][colw];
    float* O = Outp[j];
    #pragma unroll
    for (int mt = 0; mt < 2; ++mt) {
      #pragma unroll
      for (int mm = 0; mm < 8; ++mm) {
        int row = m_base + mt * 16 + mm + 8 * half;
        if (row < n) O[row * DD + colw] = acc[mt][j][mm] + bb;
      }
    }
  }
}

// =====================================================================
// Kernel 2 (pass A): per 16-edge tile recompute e = attr@We + be via
// WMMA (We staged once per block in LDS), then per-(edge,head) scores
// and segment max via encoded-uint atomicMax.
// =====================================================================
#define WE_STRIDE 132
#define ET_STRIDE 132

__global__ __launch_bounds__(256) void edge_score_kernel(
    const float* __restrict__ edge_attr,
    const int* __restrict__ src, const int* __restrict__ dst,
    const float* __restrict__ We, const float* __restrict__ be,
    const float* __restrict__ q, const float* __restrict__ k,
    float* __restrict__ score, unsigned* __restrict__ mmax, int nEdges)
{
  __shared__ float Wes[DD * WE_STRIDE];
  __shared__ float As[16 * WE_STRIDE];
  __shared__ float Es[16 * ET_STRIDE];
  __shared__ float Ps[16][16];

  const int tid  = threadIdx.x;
  const int wave = tid >> 5;
  const int lane = tid & 31;
  const int half = lane >> 4;
  const int idx  = lane & 15;

  for (int i = tid; i < DD * DD; i += 256) {
    int r = i >> 7, c = i & 127;
    Wes[r * WE_STRIDE + c] = We[i];
  }
  __syncthreads();

  const int nTiles = nEdges >> 4;
  for (int tile = blockIdx.x; tile < nTiles; tile += gridDim.x) {
    const int e0 = tile << 4;

    int nxt = tile + gridDim.x;
    if (nxt < nTiles)
      __builtin_prefetch(edge_attr + (size_t)(nxt << 4) * DD + tid * 8, 0, 0);

    for (int i = tid; i < 16 * DD; i += 256) {
      int r = i >> 7, c = i & 127;
      As[r * WE_STRIDE + c] = edge_attr[(size_t)(e0 + r) * DD + c];
    }
    __syncthreads();

    // e-tile GEMM: each wave owns one 16-wide N tile
    v8f acc = (v8f){0,0,0,0,0,0,0,0};
    const int n0 = wave * 16;
    for (int k0 = 0; k0 < DD; k0 += 4) {
      v2f a, b;
      a.x = As[idx * WE_STRIDE + k0 + 2 * half + 0];
      a.y = As[idx * WE_STRIDE + k0 + 2 * half + 1];
      b.x = Wes[(k0 + 2 * half + 0) * WE_STRIDE + n0 + idx];
      b.y = Wes[(k0 + 2 * half + 1) * WE_STRIDE + n0 + idx];
      acc = __builtin_amdgcn_wmma_f32_16x16x4_f32(
          false, a, false, b, (short)0, acc, false, false);
    }
    {
      float bb = be[n0 + idx];
      #pragma unroll
      for (int mm = 0; mm < 8; ++mm)
        Es[(mm + 8 * half) * ET_STRIDE + n0 + idx] = acc[mm] + bb;
    }
    __syncthreads();

    // scores: thread = (edge er, 8-col slice sub)
    {
      const int er  = tid >> 4;
      const int sub = tid & 15;
      const int e_id = e0 + er;
      const int s = src[e_id], d = dst[e_id];
      const int c0 = sub * 8;
      const float4* qv = reinterpret_cast<const float4*>(q + (size_t)d * DD + c0);
      const float4* kv = reinterpret_cast<const float4*>(k + (size_t)s * DD + c0);
      float4 q0 = qv[0], q1 = qv[1];
      float4 k0v = kv[0], k1v = kv[1];
      const float* es = &Es[er * ET_STRIDE + c0];
      float p = q0.x * (k0v.x + es[0]) + q0.y * (k0v.y + es[1])
              + q0.z * (k0v.z + es[2]) + q0.w * (k0v.w + es[3])
              + q1.x * (k1v.x + es[4]) + q1.y * (k1v.y + es[5])
              + q1.z * (k1v.z + es[6]) + q1.w * (k1v.w + es[7]);
      Ps[er][sub] = p;
    }
    __syncthreads();
    if (tid < 64) {
      const int er = tid >> 2, h = tid & 3;
      float sc = (Ps[er][4*h] + Ps[er][4*h+1] + Ps[er][4*h+2] + Ps[er][4*h+3])
                 * INV_SQRT_C;
      const int e_id = e0 + er;
      score[(size_t)e_id * HH + h] = sc;
      atomicMax(&mmax[(size_t)dst[e_id] * HH + h], f32_ord(sc));
    }
    __syncthreads();
  }
}

// =====================================================================
// Kernel 3 (pass B): ex = exp(score - max[dst]); den[dst] += ex
// =====================================================================
__global__ __launch_bounds__(256) void edge_exp_kernel(
    const int* __restrict__ dst, float* __restrict__ score,
    const unsigned* __restrict__ mmax, float* __restrict__ den, int total)
{
  int i = blockIdx.x * 256 + threadIdx.x;
  if (i >= total) return;
  int e = i >> 2, h = i & 3;
  int d = dst[e];
  float ex = __expf(score[i] - ord_f32(mmax[(size_t)d * HH + h]));
  score[i] = ex;
  atomicAdd(&den[(size_t)d * HH + h], ex);
}

// =====================================================================
// Kernel 4 (pass C): recompute e tile with WMMA, alpha = ex/den,
// scatter-add messages (v[src]+e)*alpha into xout (preloaded with skip).
// =====================================================================
__global__ __launch_bounds__(256) void edge_msg_kernel(
    const float* __restrict__ edge_attr,
    const int* __restrict__ src, const int* __restrict__ dst,
    const float* __restrict__ We, const float* __restrict__ be,
    const float* __restrict__ v, const float* __restrict__ score,
    const float* __restrict__ den, float* __restrict__ xout, int nEdges)
{
  __shared__ float Wes[DD * WE_STRIDE];
  __shared__ float As[16 * WE_STRIDE];
  __shared__ float Es[16 * ET_STRIDE];

  const int tid  = threadIdx.x;
  const int wave = tid >> 5;
  const int lane = tid & 31;
  const int half = lane >> 4;
  const int idx  = lane & 15;

  for (int i = tid; i < DD * DD; i += 256) {
    int r = i >> 7, c = i & 127;
    Wes[r * WE_STRIDE + c] = We[i];
  }
  __syncthreads();

  const int nTiles = nEdges >> 4;
  for (int tile = blockIdx.x; tile < nTiles; tile += gridDim.x) {
    const int e0 = tile << 4;

    int nxt = tile + gridDim.x;
    if (nxt < nTiles)
      __builtin_prefetch(edge_attr + (size_t)(nxt << 4) * DD + tid * 8, 0, 0);

    for (int i = tid; i < 16 * DD; i += 256) {
      int r = i >> 7, c = i & 127;
      As[r * WE_STRIDE + c] = edge_attr[(size_t)(e0 + r) * DD + c];
    }
    __syncthreads();

    v8f acc = (v8f){0,0,0,0,0,0,0,0};
    const int n0 = wave * 16;
    for (int k0 = 0; k0 < DD; k0 += 4) {
      v2f a, b;
      a.x = As[idx * WE_STRIDE + k0 + 2 * half + 0];
      a.y = As[idx * WE_STRIDE + k0 + 2 * half + 1];
      b.x = Wes[(k0 + 2 * half + 0) * WE_STRIDE + n0 + idx];
      b.y = Wes[(k0 + 2 * half + 1) * WE_STRIDE + n0 + idx];
      acc = __builtin_amdgcn_wmma_f32_16x16x4_f32(
          false, a, false, b, (short)0, acc, false, false);
    }
    {
      float bb = be[n0 + idx];
      #pragma unroll
      for (int mm = 0; mm < 8; ++mm)
        Es[(mm + 8 * half) * ET_STRIDE + n0 + idx] = acc[mm] + bb;
    }
    __syncthreads();

    {
      const int er  = tid >> 4;
      const int sub = tid & 15;
      const int e_id = e0 + er;
      const int s = src[e_id], d = dst[e_id];
      const int h = sub >> 2;                    // head of this 8-col slice
      const int c0 = sub * 8;
      float alpha = score[(size_t)e_id * HH + h] / den[(size_t)d * HH + h];
      const float4* vv = reinterpret_cast<const float4*>(v + (size_t)s * DD + c0);
      float4 v0 = vv[0], v1 = vv[1];
      const float* es = &Es[er * ET_STRIDE + c0];
      float* out = xout + (size_t)d * DD + c0;
      atomicAdd(out + 0, (v0.x + es[0]) * alpha);
      atomicAdd(out + 1, (v0.y + es[1]) * alpha);
      atomicAdd(out + 2, (v0.z + es[2]) * alpha);
      atomicAdd(out + 3, (v0.w + es[3]) * alpha);
      atomicAdd(out + 4, (v1.x + es[4]) * alpha);
      atomicAdd(out + 5, (v1.y + es[5]) * alpha);
      atomicAdd(out + 6, (v1.z + es[6]) * alpha);
      atomicAdd(out + 7, (v1.w + es[7]) * alpha);
    }
    __syncthreads();
  }
}

// =====================================================================
// BN: per-column sum / sumsq, then normalize + ReLU in place.
// =====================================================================
__global__ __launch_bounds__(256) void bn_stats_kernel(
    const float* __restrict__ x, float* __restrict__ stats, int n)
{
  const int col = threadIdx.x & 127;
  const int rg  = threadIdx.x >> 7;        // 0 or 1
  const int base = blockIdx.x * 128;
  float s = 0.f, s2 = 0.f;
  for (int r = base + rg; r < base + 128 && r < n; r += 2) {
    float xv = x[(size_t)r * DD + col];
    s += xv; s2 += xv * xv;
  }
  atomicAdd(&stats[col], s);
  atomicAdd(&stats[DD + col], s2);
}

__global__ __launch_bounds__(256) void bn_apply_kernel(
    float* __restrict__ x, const float* __restrict__ stats,
    const float* __restrict__ gamma, const float* __restrict__ beta, int n)
{
  int i = blockIdx.x * 256 + threadIdx.x;
  if (i >= n * DD) return;
  int col = i & 127;
  float inv_n = 1.0f / (float)n;
  float mu  = stats[col] * inv_n;
  float var = stats[DD + col] * inv_n - mu * mu;
  float y = gamma[col] * (x[i] - mu) * rsqrtf(var + BN_EPS) + beta[col];
  x[i] = fmaxf(y, 0.0f);
}

// =====================================================================
// Host launcher
// =====================================================================
extern "C" void kernel_launch(void* const* d_in, const int* in_sizes, int n_in,
                              void* d_out, int out_size, void* d_ws, size_t ws_size,
                              hipStream_t stream)
{
  const float* x0        = (const float*)d_in[0];
  const float* edge_attr = (const float*)d_in[1];
  const int*   eidx      = (const int*)d_in[2];       // int32 (JAX x64 off)
  const int*   src = eidx;
  const int*   dst = eidx + N_EDGES;

  // Workspace carve-up (floats)
  float* ws = (float*)d_ws;
  float* q   = ws;                       // N*D
  float* kk  = q  + (size_t)N_NODES * DD;
  float* vv  = kk + (size_t)N_NODES * DD;
  float* xA  = vv + (size_t)N_NODES * DD;
  float* xB  = xA + (size_t)N_NODES * DD;
  float* score = xB + (size_t)N_NODES * DD;           // E*H
  unsigned* mmax = (unsigned*)(score + (size_t)N_EDGES * HH);  // N*H
  float* den   = (float*)(mmax + (size_t)N_NODES * HH);        // N*H
  float* stats = den + (size_t)N_NODES * HH;                   // 2*D
  // mmax..stats contiguous zero region:
  const size_t zero_bytes = ((size_t)2 * N_NODES * HH + 2 * DD) * sizeof(float);

  for (int l = 0; l < 3; ++l) {
    const int pb = 3 + 12 * l;
    const float* Wq = (const float*)d_in[pb + 0];
    const float* bq = (const float*)d_in[pb + 1];
    const float* Wk = (const float*)d_in[pb + 2];
    const float* bk = (const float*)d_in[pb + 3];
    const float* Wv = (const float*)d_in[pb + 4];
    const float* bv = (const float*)d_in[pb + 5];
    const float* We = (const float*)d_in[pb + 6];
    const float* be = (const float*)d_in[pb + 7];
    const float* Wsk = (const float*)d_in[pb + 8];
    const float* bsk = (const float*)d_in[pb + 9];
    const float* gamma = (const float*)d_in[pb + 10];
    const float* beta  = (const float*)d_in[pb + 11];

    const float* xin = (l == 0) ? x0 : ((l == 1) ? xA : xB);
    float* xout = (l == 0) ? xA : ((l == 1) ? xB : (float*)d_out);

    hipMemsetAsync(mmax, 0, zero_bytes, stream);

    node_qkvs_kernel<<<(N_NODES + 31) / 32, 256, 0, stream>>>(
        xin, Wq, bq, Wk, bk, Wv, bv, Wsk, bsk, q, kk, vv, xout, N_NODES);

    edge_score_kernel<<<1280, 256, 0, stream>>>(
        edge_attr, src, dst, We, be, q, kk, score, mmax, N_EDGES);

    edge_exp_kernel<<<(N_EDGES * HH + 255) / 256, 256, 0, stream>>>(
        dst, score, mmax, den, N_EDGES * HH);

    edge_msg_kernel<<<1280, 256, 0, stream>>>(
        edge_attr, src, dst, We, be, vv, score, den, xout, N_EDGES);

    if (l < 2) {
      bn_stats_kernel<<<(N_NODES + 127) / 128, 256, 0, stream>>>(
          xout, stats, N_NODES);
      bn_apply_kernel<<<((size_t)N_NODES * DD + 255) / 256, 256, 0, stream>>>(
          xout, stats, gamma, beta, N_NODES);
    }
  }
}